// Encoder_89318139888059
// MI455X (gfx1250) — compile-verified
//
#include <hip/hip_runtime.h>
#include <math.h>

#define BB     8192
#define BAND   5
#define NN     62
#define CBAMC  15
#define HID    3
#define H2     64
#define KTOP   49
#define NF     930
#define KPAD   960

// workspace layout (floats)
#define WS_FEAT_OFF   0
#define WS_FEAT_LEN   ((size_t)BB * KPAD)                 // 8192*960
#define WS_B1_OFF     (WS_FEAT_OFF + WS_FEAT_LEN)
#define WS_B1_LEN     ((size_t)(KPAD / 2) * H2 * 2)       // 480*128 = 61440
#define WS_B2_OFF     (WS_B1_OFF + WS_B1_LEN)
#define WS_B2_LEN     ((size_t)(H2 / 2) * H2 * 2)         // 32*128  = 4096

typedef __attribute__((ext_vector_type(2))) float v2f;
typedef __attribute__((ext_vector_type(8))) float v8f;

#define V8ZERO {0.f,0.f,0.f,0.f,0.f,0.f,0.f,0.f}

__device__ __forceinline__ float sigmoidf_(float v) { return 1.f / (1.f + __expf(-v)); }

// ---------------------------------------------------------------------------
// Kernel 0: pack fc1_w / fc2_w into WMMA B-fragment pair layout.
//   B1p[p*128 + n*2 + j] = (2p+j < 930) ? fc1_w[n*930 + 2p + j] : 0
//   B2p[p*128 + n*2 + j] =                fc2_w[n*64  + 2p + j]
// ---------------------------------------------------------------------------
__global__ __launch_bounds__(256) void pack_weights_kernel(
    const float* __restrict__ fc1w, const float* __restrict__ fc2w,
    float* __restrict__ b1p, float* __restrict__ b2p)
{
    const int idx = blockIdx.x * 256 + threadIdx.x;
    if (idx < (int)WS_B1_LEN) {
        int p = idx >> 7;            // pair index (k/2)
        int r = idx & 127;
        int n = r >> 1;
        int j = r & 1;
        int k = p * 2 + j;
        b1p[idx] = (k < NF) ? fc1w[(size_t)n * NF + k] : 0.f;
    } else {
        int t = idx - (int)WS_B1_LEN;
        if (t < (int)WS_B2_LEN) {
            int p = t >> 7;
            int r = t & 127;
            int n = r >> 1;
            int j = r & 1;
            int k = p * 2 + j;
            b2p[t] = fc2w[n * H2 + k];
        }
    }
}

// ---------------------------------------------------------------------------
// Kernel 1: one workgroup (64 threads) per batch sample.
// Runs both DGCN layers + CBAM entirely in LDS, writes the 930-vector
// (zero-padded to 960) to workspace for the WMMA FC kernel.
// ---------------------------------------------------------------------------
__global__ __launch_bounds__(64) void dgcn_cbam_kernel(
    const float* __restrict__ x,
    const float* __restrict__ conv_w0, const float* __restrict__ conv_b0,
    const float* __restrict__ mem0,    const float* __restrict__ gfc_w0,
    const float* __restrict__ gfc_b0,  const float* __restrict__ gcn_w0,
    const float* __restrict__ gcn_b0,
    const float* __restrict__ conv_w1, const float* __restrict__ conv_b1,
    const float* __restrict__ mem1,    const float* __restrict__ gfc_w1,
    const float* __restrict__ gfc_b1,  const float* __restrict__ gcn_w1,
    const float* __restrict__ gcn_b1,
    const float* __restrict__ ca_w1,   const float* __restrict__ ca_w2,
    const float* __restrict__ sa_w,    const float* __restrict__ sa_b,
    float* __restrict__ gout)
{
    __shared__ float xcur[BAND][NN];
    __shared__ float xp[BAND][NN];
    __shared__ float ynew[BAND][NN];
    __shared__ float g[CBAMC][NN];
    __shared__ float go[CBAMC][NN];
    __shared__ float af[NN][NN];
    __shared__ float dmx[CBAMC], dav[CBAMC];
    __shared__ float hmx[HID], hav[HID], cav[CBAMC];
    __shared__ float smaxA[NN], smeanA[NN], saA[NN];

    const int b = blockIdx.x;
    const int tid = threadIdx.x;
    const float scale = 0.4472135954999579f;   // 1/sqrt(BAND)

    for (int i = tid; i < BAND * NN; i += 64) {
        int c = i / NN, n = i % NN;
        float v = x[(size_t)b * (BAND * NN) + i];
        xcur[c][n] = v;
        g[c][n] = v;          // feats[0]
    }
    __syncthreads();

    for (int l = 0; l < 2; ++l) {
        const float* convw = l ? conv_w1 : conv_w0;
        const float* convb = l ? conv_b1 : conv_b0;
        const float* memw  = l ? mem1    : mem0;
        const float* gfcw  = l ? gfc_w1  : gfc_w0;
        const float* gfcb  = l ? gfc_b1  : gfc_b0;
        const float* gcnw  = l ? gcn_w1  : gcn_w0;
        const float* gcnb  = l ? gcn_b1  : gcn_b0;

        // 1x1 conv over channels
        for (int i = tid; i < BAND * NN; i += 64) {
            int c = i / NN, n = i % NN;
            float s = convb[c];
            #pragma unroll
            for (int c2 = 0; c2 < BAND; ++c2) s += convw[c * BAND + c2] * xcur[c2][n];
            xp[c][n] = s;
        }
        __syncthreads();

        // adjacency: one thread per row n
        if (tid < NN) {
            const int n = tid;
            float xn[BAND];
            #pragma unroll
            for (int c = 0; c < BAND; ++c) xn[c] = xp[c][n];

            // a1 = softmax(relu(xp . mem * scale)) -> af row holds exp values
            float mx1 = -3.4e38f;
            for (int m = 0; m < NN; ++m) {
                float v = 0.f;
                #pragma unroll
                for (int c = 0; c < BAND; ++c) v += xn[c] * memw[c * NN + m];
                v = fmaxf(v * scale, 0.f);
                af[n][m] = v; mx1 = fmaxf(mx1, v);
            }
            float s1 = 0.f;
            for (int m = 0; m < NN; ++m) { float e = __expf(af[n][m] - mx1); af[n][m] = e; s1 += e; }
            float inv1 = 1.f / s1;

            // a2 = softmax(relu(xs . xs * scale)); recompute cheap 5-MAC dots
            float mx2 = -3.4e38f;
            for (int m = 0; m < NN; ++m) {
                float v = 0.f;
                #pragma unroll
                for (int c = 0; c < BAND; ++c) v += xn[c] * xp[c][m];
                v = fmaxf(v * scale, 0.f);
                mx2 = fmaxf(mx2, v);
            }
            float s2 = 0.f;
            for (int m = 0; m < NN; ++m) {
                float v = 0.f;
                #pragma unroll
                for (int c = 0; c < BAND; ++c) v += xn[c] * xp[c][m];
                v = fmaxf(v * scale, 0.f);
                s2 += __expf(v - mx2);
            }
            float inv2 = 1.f / s2;

            // combine via gfc (2->1), then softmax again
            float w0 = gfcw[0], w1 = gfcw[1], gb = gfcb[0];
            float mx3 = -3.4e38f;
            for (int m = 0; m < NN; ++m) {
                float v = 0.f;
                #pragma unroll
                for (int c = 0; c < BAND; ++c) v += xn[c] * xp[c][m];
                v = fmaxf(v * scale, 0.f);
                float a2v = __expf(v - mx2) * inv2;
                float u = w0 * (af[n][m] * inv1) + w1 * a2v + gb;
                af[n][m] = u; mx3 = fmaxf(mx3, u);
            }
            float s3 = 0.f;
            for (int m = 0; m < NN; ++m) { float e = __expf(af[n][m] - mx3); af[n][m] = e; s3 += e; }
            float inv3 = 1.f / s3;
            for (int m = 0; m < NN; ++m) af[n][m] *= inv3;

            // top-49 mask per row (rank with earlier-index tie-break = top_k)
            unsigned keep0 = 0u, keep1 = 0u;
            for (int m = 0; m < NN; ++m) {
                float v = af[n][m];
                int rank = 0;
                for (int mp = 0; mp < NN; ++mp) {
                    float w = af[n][mp];
                    rank += (w > v) || ((w == v) && (mp < m));
                }
                if (rank < KTOP) { if (m < 32) keep0 |= 1u << m; else keep1 |= 1u << (m - 32); }
            }
            for (int m = 0; m < NN; ++m) {
                bool k = (m < 32) ? ((keep0 >> m) & 1u) : ((keep1 >> (m - 32)) & 1u);
                if (!k) af[n][m] = 0.f;
            }
        }
        __syncthreads();

        // diffusion: y[c][m] = sum_n xp[c][n] * adj[n][m]
        for (int i = tid; i < BAND * NN; i += 64) {
            int c = i / NN, m = i % NN;
            float s = 0.f;
            for (int n2 = 0; n2 < NN; ++n2) s += xp[c][n2] * af[n2][m];
            ynew[c][m] = s;
        }
        __syncthreads();

        // gcn 1x1 conv + skip
        for (int i = tid; i < BAND * NN; i += 64) {
            int c = i / NN, n = i % NN;
            float s = gcnb[c];
            #pragma unroll
            for (int c2 = 0; c2 < BAND; ++c2) s += gcnw[c * BAND + c2] * ynew[c2][n];
            s += xcur[c][n];
            xcur[c][n] = s;
            g[(l + 1) * BAND + c][n] = s;
        }
        __syncthreads();
    }

    // ---- CBAM: channel attention ----
    if (tid < CBAMC) {
        float mx = -3.4e38f, sm = 0.f;
        for (int n = 0; n < NN; ++n) { float v = g[tid][n]; mx = fmaxf(mx, v); sm += v; }
        dmx[tid] = mx; dav[tid] = sm * (1.f / NN);
    }
    __syncthreads();
    if (tid < HID) {
        float h1 = 0.f, h2 = 0.f;
        for (int c = 0; c < CBAMC; ++c) {
            h1 += ca_w1[tid * CBAMC + c] * dmx[c];
            h2 += ca_w1[tid * CBAMC + c] * dav[c];
        }
        hmx[tid] = fmaxf(h1, 0.f); hav[tid] = fmaxf(h2, 0.f);
    }
    __syncthreads();
    if (tid < CBAMC) {
        float s = 0.f;
        for (int h = 0; h < HID; ++h) s += ca_w2[tid * HID + h] * (hmx[h] + hav[h]);
        cav[tid] = sigmoidf_(s);
    }
    __syncthreads();
    for (int i = tid; i < CBAMC * NN; i += 64) {
        int c = i / NN, n = i % NN;
        go[c][n] = g[c][n] * cav[c];
    }
    __syncthreads();

    // ---- CBAM: spatial attention (3x3 conv, W=1 -> only center column) ----
    if (tid < NN) {
        float mx = -3.4e38f, sm = 0.f;
        for (int c = 0; c < CBAMC; ++c) { float v = go[c][tid]; mx = fmaxf(mx, v); sm += v; }
        smaxA[tid] = mx; smeanA[tid] = sm * (1.f / CBAMC);
    }
    __syncthreads();
    if (tid < NN) {
        float acc = sa_b[0];
        #pragma unroll
        for (int kh = 0; kh < 3; ++kh) {
            int j = tid + kh - 1;
            if (j >= 0 && j < NN)
                acc += smaxA[j] * sa_w[kh * 3 + 1] + smeanA[j] * sa_w[9 + kh * 3 + 1];
        }
        saA[tid] = sigmoidf_(acc);
    }
    __syncthreads();

    // final = go*sa + g, flattened c*62+n, zero-padded to 960
    float* dst = gout + (size_t)b * KPAD;
    for (int i = tid; i < KPAD; i += 64) {
        float v = 0.f;
        if (i < CBAMC * NN) {
            int c = i / NN, n = i % NN;
            v = go[c][n] * saA[n] + g[c][n];
        }
        dst[i] = v;
    }
}

// ---------------------------------------------------------------------------
// Kernel 2: batched FC layers with fp32 WMMA (V_WMMA_F32_16X16X4_F32).
// 8 waves per block; each wave owns 16 batch rows and all 64 output cols.
// B matrices come pre-packed in pair layout -> branch-free b64 loads.
// ---------------------------------------------------------------------------
__global__ __launch_bounds__(256) void fc_wmma_kernel(
    const float* __restrict__ gin,
    const float* __restrict__ b1p, const float* __restrict__ fc1b,
    const float* __restrict__ b2p, const float* __restrict__ fc2b,
    float* __restrict__ out)
{
    __shared__ float h1[128][H2 + 2];   // +2 padding: avoid LDS bank conflicts

    const int lane = threadIdx.x & 31;
    const int wv   = threadIdx.x >> 5;
    const int row0 = blockIdx.x * 128 + wv * 16;   // this wave's 16 batch rows
    const int m    = lane & 15;                    // M (A) / N (B,D) index
    const int kh   = lane >> 4;                    // K-half select

    // ---- fc1: [16 x 960] x [960 x 64] ----
    v8f acc[4] = { V8ZERO, V8ZERO, V8ZERO, V8ZERO };
    const float* aBase = gin + (size_t)(row0 + m) * KPAD + 2 * kh;     // 8B aligned
    #pragma unroll 2
    for (int k = 0; k < KPAD; k += 4) {
        __builtin_prefetch(aBase + k + 128, 0, 1);
        v2f a = *(const v2f*)(aBase + k);
        const int p = (k >> 1) + kh;               // pair index into packed B
        const float* bRow = b1p + (size_t)p * (H2 * 2) + 2 * m;
        #pragma unroll
        for (int nt = 0; nt < 4; ++nt) {
            v2f bb = *(const v2f*)(bRow + nt * 32);
            acc[nt] = __builtin_amdgcn_wmma_f32_16x16x4_f32(
                false, a, false, bb, (short)0, acc[nt], false, false);
        }
    }
    // bias + relu -> LDS (D layout: vgpr r, lanes<16 -> M=r, lanes>=16 -> M=r+8)
    #pragma unroll
    for (int nt = 0; nt < 4; ++nt) {
        const int col = nt * 16 + m;
        const float bias = fc1b[col];
        #pragma unroll
        for (int r = 0; r < 8; ++r) {
            const int rl = wv * 16 + r + 8 * kh;
            h1[rl][col] = fmaxf(acc[nt][r] + bias, 0.f);
        }
    }
    __syncthreads();

    // ---- fc2: [16 x 64] x [64 x 64] ----
    v8f acc2[4] = { V8ZERO, V8ZERO, V8ZERO, V8ZERO };
    #pragma unroll 2
    for (int k = 0; k < H2; k += 4) {
        const int kb = k + 2 * kh;
        v2f a; a.x = h1[wv * 16 + m][kb]; a.y = h1[wv * 16 + m][kb + 1];
        const int p = (k >> 1) + kh;
        const float* bRow = b2p + (size_t)p * (H2 * 2) + 2 * m;
        #pragma unroll
        for (int nt = 0; nt < 4; ++nt) {
            v2f bb = *(const v2f*)(bRow + nt * 32);
            acc2[nt] = __builtin_amdgcn_wmma_f32_16x16x4_f32(
                false, a, false, bb, (short)0, acc2[nt], false, false);
        }
    }
    #pragma unroll
    for (int nt = 0; nt < 4; ++nt) {
        const int col = nt * 16 + m;
        const float bias = fc2b[col];
        #pragma unroll
        for (int r = 0; r < 8; ++r) {
            const int row = row0 + r + 8 * kh;
            out[(size_t)row * H2 + col] = fmaxf(acc2[nt][r] + bias, 0.f);
        }
    }
}

// ---------------------------------------------------------------------------
extern "C" void kernel_launch(void* const* d_in, const int* in_sizes, int n_in,
                              void* d_out, int out_size, void* d_ws, size_t ws_size,
                              hipStream_t stream) {
    (void)in_sizes; (void)n_in; (void)out_size; (void)ws_size;

    const float* x        = (const float*)d_in[0];
    const float* l0_cw    = (const float*)d_in[1];
    const float* l0_cb    = (const float*)d_in[2];
    const float* l0_mem   = (const float*)d_in[3];
    const float* l0_gfw   = (const float*)d_in[4];
    const float* l0_gfb   = (const float*)d_in[5];
    const float* l0_gw    = (const float*)d_in[6];
    const float* l0_gb    = (const float*)d_in[7];
    const float* l1_cw    = (const float*)d_in[8];
    const float* l1_cb    = (const float*)d_in[9];
    const float* l1_mem   = (const float*)d_in[10];
    const float* l1_gfw   = (const float*)d_in[11];
    const float* l1_gfb   = (const float*)d_in[12];
    const float* l1_gw    = (const float*)d_in[13];
    const float* l1_gb    = (const float*)d_in[14];
    const float* ca_w1    = (const float*)d_in[15];
    const float* ca_w2    = (const float*)d_in[16];
    const float* sa_w     = (const float*)d_in[17];
    const float* sa_b     = (const float*)d_in[18];
    const float* fc1_w    = (const float*)d_in[19];
    const float* fc1_b    = (const float*)d_in[20];
    const float* fc2_w    = (const float*)d_in[21];
    const float* fc2_b    = (const float*)d_in[22];

    float* ws   = (float*)d_ws;
    float* gws  = ws + WS_FEAT_OFF;    // [B, 960] padded feature vectors
    float* b1p  = ws + WS_B1_OFF;      // packed fc1 B fragments
    float* b2p  = ws + WS_B2_OFF;      // packed fc2 B fragments

    pack_weights_kernel<<<(int)((WS_B1_LEN + WS_B2_LEN + 255) / 256), 256, 0, stream>>>(
        fc1_w, fc2_w, b1p, b2p);

    dgcn_cbam_kernel<<<BB, 64, 0, stream>>>(
        x,
        l0_cw, l0_cb, l0_mem, l0_gfw, l0_gfb, l0_gw, l0_gb,
        l1_cw, l1_cb, l1_mem, l1_gfw, l1_gfb, l1_gw, l1_gb,
        ca_w1, ca_w2, sa_w, sa_b, gws);

    fc_wmma_kernel<<<BB / 128, 256, 0, stream>>>(
        gws, b1p, fc1_b, b2p, fc2_b, (float*)d_out);
}